// HatNet_22608707846558
// MI455X (gfx1250) — compile-verified
//
#include <hip/hip_runtime.h>
#include <math.h>

#define B_   128
#define N_   548
#define K1_  274
#define K2_  137

typedef __attribute__((ext_vector_type(2))) float v2f;
typedef __attribute__((ext_vector_type(8))) float v8f;

#define CDIV(a,b) (((a)+(b)-1)/(b))

// ---------------------------------------------------------------- reductions
static __device__ __forceinline__ float warpReduceSum(float v){
#pragma unroll
  for (int o = 16; o > 0; o >>= 1) v += __shfl_xor(v, o, 32);
  return v;
}
static __device__ float blockReduceSum(float v){
  __shared__ float sh[32];
  int lane = threadIdx.x & 31, wid = threadIdx.x >> 5;
  v = warpReduceSum(v);
  __syncthreads();
  if (lane == 0) sh[wid] = v;
  __syncthreads();
  int nw = (blockDim.x + 31) >> 5;
  v = (threadIdx.x < (unsigned)nw) ? sh[threadIdx.x] : 0.0f;
  if (wid == 0) v = warpReduceSum(v);
  return v;  // valid on thread 0
}

// ---------------------------------------------------------------- elementwise
__global__ void k_zero(float* p, long n){
  for (long i = (long)blockIdx.x*blockDim.x + threadIdx.x; i < n; i += (long)gridDim.x*blockDim.x)
    p[i] = 0.0f;
}
__global__ void k_copy(const float* __restrict__ s, float* __restrict__ d, long n){
  for (long i = (long)blockIdx.x*blockDim.x + threadIdx.x; i < n; i += (long)gridDim.x*blockDim.x)
    d[i] = s[i];
}
// acc[dst,:] += x[src,:]  (acc pre-initialized with x -> yields x + aggregate)
__global__ void k_scatter(const float* __restrict__ x, float* acc,
                          const int* __restrict__ src, const int* __restrict__ dst, long E){
  long total = E << 6;
  for (long i = (long)blockIdx.x*blockDim.x + threadIdx.x; i < total; i += (long)gridDim.x*blockDim.x){
    long e = i >> 6; int c = (int)(i & 63);
    int s = src[e], d = dst[e];
    atomicAdd(&acc[(long)d*64 + c], x[(long)s*64 + c]);
  }
}
__global__ void k_adj(float* adj, const int* __restrict__ src, const int* __restrict__ dst, long E, int Nn){
  for (long e = (long)blockIdx.x*blockDim.x + threadIdx.x; e < E; e += (long)gridDim.x*blockDim.x){
    int s = src[e], d = dst[e];
    int g = s / Nn;
    atomicAdd(&adj[(long)g*Nn*Nn + (long)(s % Nn)*Nn + (d % Nn)], 1.0f);
  }
}
// out[i] = sum_j mat[i*cols + j], i in [0, total)  (batched row sums)
__global__ void k_rowsum(const float* __restrict__ mat, float* out, int cols, long total){
  for (long i = (long)blockIdx.x*blockDim.x + threadIdx.x; i < total; i += (long)gridDim.x*blockDim.x){
    const float* p = mat + i*(long)cols;
    float s = 0.0f;
    for (int j = 0; j < cols; ++j) s += p[j];
    out[i] = s;
  }
}
__global__ void k_bn_prep(const float* g, const float* be, const float* rm, const float* rv,
                          float* scale, float* shift, int n){
  int i = threadIdx.x;
  if (i < n){
    float sc = g[i] * rsqrtf(rv[i] + 1e-5f);
    scale[i] = sc;
    shift[i] = be[i] - rm[i]*sc;
  }
}
// one wave per row softmax
__global__ void k_softmax(float* s, long rows, int Kc){
  long row = (long)blockIdx.x*(blockDim.x >> 5) + (threadIdx.x >> 5);
  int lane = threadIdx.x & 31;
  if (row >= rows) return;
  float* p = s + row*(long)Kc;
  float mx = -1e30f;
  for (int k = lane; k < Kc; k += 32) mx = fmaxf(mx, p[k]);
#pragma unroll
  for (int o = 16; o > 0; o >>= 1) mx = fmaxf(mx, __shfl_xor(mx, o, 32));
  float sum = 0.0f;
  for (int k = lane; k < Kc; k += 32){ float e = expf(p[k]-mx); p[k] = e; sum += e; }
#pragma unroll
  for (int o = 16; o > 0; o >>= 1) sum += __shfl_xor(sum, o, 32);
  float inv = 1.0f / sum;
  for (int k = lane; k < Kc; k += 32) p[k] *= inv;
}
// mincut loss contribution for one batch per block (before diag zeroing)
__global__ void k_mincut(const float* __restrict__ oadj, const float* __restrict__ s,
                         const float* __restrict__ deg, int Nn, int Kc, float* out, float invB){
  int b = blockIdx.x;
  const float* oa = oadj + (long)b*Kc*Kc;
  float num = 0.0f;
  for (int k = threadIdx.x; k < Kc; k += blockDim.x) num += oa[(long)k*Kc + k];
  num = blockReduceSum(num);
  __shared__ float sNum;
  if (threadIdx.x == 0) sNum = num;
  __syncthreads();
  float den = 0.0f;
  long tot = (long)Nn*Kc;
  for (long i = threadIdx.x; i < tot; i += blockDim.x){
    int n = (int)(i / Kc);
    float sv = s[((long)b*Nn + n)*Kc + (i % Kc)];
    den += deg[(long)b*Nn + n]*sv*sv;
  }
  den = blockReduceSum(den);
  if (threadIdx.x == 0) atomicAdd(out, -(sNum/den)*invB);
}
__global__ void k_ortho(const float* __restrict__ ss, int Kc, float* out, float invB){
  int b = blockIdx.x;
  const float* p = ss + (long)b*Kc*Kc;
  long tot = (long)Kc*Kc;
  float s1 = 0.0f;
  for (long i = threadIdx.x; i < tot; i += blockDim.x){ float v = p[i]; s1 += v*v; }
  s1 = blockReduceSum(s1);
  __shared__ float shn;
  if (threadIdx.x == 0) shn = sqrtf(s1);
  __syncthreads();
  float ssn = shn, rk = rsqrtf((float)Kc);
  float s2 = 0.0f;
  for (long i = threadIdx.x; i < tot; i += blockDim.x){
    int r = (int)(i / Kc), c = (int)(i % Kc);
    float d = p[i]/ssn - ((r == c) ? rk : 0.0f);
    s2 += d*d;
  }
  s2 = blockReduceSum(s2);
  if (threadIdx.x == 0) atomicAdd(out, sqrtf(s2)*invB);
}
// zero diag, dd[k] = sqrt(rowsum) + eps  (one batch per block; one row per thread)
__global__ void k_diag_dd(float* oadj, float* dd, int Kc){
  int b = blockIdx.x;
  float* oa = oadj + (long)b*Kc*Kc;
  for (int k = threadIdx.x; k < Kc; k += blockDim.x){
    oa[(long)k*Kc + k] = 0.0f;
    float s = 0.0f;
    for (int j = 0; j < Kc; ++j) s += oa[(long)k*Kc + j];
    dd[(long)b*Kc + k] = sqrtf(s) + 1e-15f;
  }
}
__global__ void k_normalize(float* oadj, const float* __restrict__ dd, int Kc, long total){
  long kk = (long)Kc*Kc;
  for (long i = (long)blockIdx.x*blockDim.x + threadIdx.x; i < total; i += (long)gridDim.x*blockDim.x){
    long b = i / kk;
    int r = (int)((i / Kc) % Kc), c = (int)(i % Kc);
    oadj[i] /= (dd[b*Kc + r]*dd[b*Kc + c]);
  }
}
__global__ void k_mean(const float* __restrict__ x, float* h, int Kc){
  int tid = blockIdx.x*blockDim.x + threadIdx.x;
  if (tid >= B_*64) return;
  int b = tid >> 6, c = tid & 63;
  float s = 0.0f;
  for (int n = 0; n < Kc; ++n) s += x[((long)b*Kc + n)*64 + c];
  h[tid] = s / (float)Kc;
}

// ---------------------------------------------------------------- WMMA GEMM
// C[b](MxN) = op(A[b]) * B[b]  [+bias][*scale+shift][relu][+Cadd]
// transA: stored A is (K x M) with row stride lda, op(A)(m,k) = A[k*lda + m]
// One wave computes a 16x32 output tile (2 accumulators share the A fragment).
// OOB lanes use clamped (always-legal) addresses + cndmask-zero: no divergent
// branches in the K loop, EXEC stays all-ones for every WMMA.
__global__ void __launch_bounds__(128)
k_gemm(const float* __restrict__ A, const float* __restrict__ Bm,
       float* __restrict__ C, const float* __restrict__ Cadd,
       const float* __restrict__ bias, const float* __restrict__ scale,
       const float* __restrict__ shift, int relu, int transA,
       int M, int Nc, int K, int lda, int ldb, int ldc,
       long sA, long sB, long sC)
{
  int b = blockIdx.z;
  const float* Ab = A + (long)b*sA;
  const float* Bb = Bm + (long)b*sB;
  float* Cb = C + (long)b*sC;
  const float* Cab = Cadd ? (Cadd + (long)b*sC) : nullptr;

  int wave = threadIdx.x >> 5, lane = threadIdx.x & 31;
  int m0 = (blockIdx.x*4 + wave)*16;
  int n0 = blockIdx.y*32;
  if (m0 >= M) return;                        // wave-uniform exit

  int laneM  = lane & 15;
  int laneHi = lane >> 4;                     // 0 or 1
  int kOff   = laneHi << 1;

  int row  = m0 + laneM;       bool rowOK  = row  < M;
  int col0 = n0 + laneM;       bool colOK0 = col0 < Nc;
  int col1 = n0 + 16 + laneM;  bool colOK1 = col1 < Nc;
  int rowC  = rowOK  ? row  : (M  - 1);
  int colC0 = colOK0 ? col0 : (Nc - 1);
  int colC1 = colOK1 ? col1 : (Nc - 1);
  bool doN1 = (n0 + 16) < Nc;                 // wave-uniform

  const float* Ap; long aStep;
  if (!transA){ Ap = Ab + (long)rowC*lda; aStep = 1;   }
  else        { Ap = Ab + rowC;           aStep = lda; }
  const float* Bp0 = Bb + colC0;
  const float* Bp1 = Bb + colC1;

  v8f acc0 = {}, acc1 = {};
  int Kfull = K & ~3;

  for (int k0 = 0; k0 < Kfull; k0 += 4){
    if ((k0 & 15) == 0 && (k0 + 16) < Kfull){           // uniform prefetch of next chunk
      __builtin_prefetch(Ap  + (long)(k0+16)*aStep, 0, 1);
      __builtin_prefetch(Bp0 + (long)(k0+16)*ldb,   0, 1);
    }
    int ka = k0 + kOff;
    float a0 = Ap[(long)ka*aStep];
    float a1 = Ap[(long)(ka+1)*aStep];
    v2f a;  a.x = rowOK ? a0 : 0.0f;  a.y = rowOK ? a1 : 0.0f;

    float b0 = Bp0[(long)ka*ldb];
    float b1 = Bp0[(long)(ka+1)*ldb];
    v2f bf; bf.x = colOK0 ? b0 : 0.0f; bf.y = colOK0 ? b1 : 0.0f;
    acc0 = __builtin_amdgcn_wmma_f32_16x16x4_f32(false, a, false, bf,
                                                 (short)0, acc0, false, false);
    if (doN1){
      float c0 = Bp1[(long)ka*ldb];
      float c1 = Bp1[(long)(ka+1)*ldb];
      v2f bg; bg.x = colOK1 ? c0 : 0.0f; bg.y = colOK1 ? c1 : 0.0f;
      acc1 = __builtin_amdgcn_wmma_f32_16x16x4_f32(false, a, false, bg,
                                                   (short)0, acc1, false, false);
    }
  }
  if (Kfull < K){                                       // K tail (1..3)
    int ka = Kfull + kOff;
    bool k0ok = ka < K, k1ok = (ka+1) < K;
    long kc0 = k0ok ? ka : (K-1);
    long kc1 = k1ok ? (ka+1) : (K-1);
    float a0 = Ap[kc0*aStep], a1 = Ap[kc1*aStep];
    v2f a;  a.x = (rowOK && k0ok) ? a0 : 0.0f;
            a.y = (rowOK && k1ok) ? a1 : 0.0f;
    float b0 = Bp0[kc0*ldb], b1 = Bp0[kc1*ldb];
    v2f bf; bf.x = (colOK0 && k0ok) ? b0 : 0.0f;
            bf.y = (colOK0 && k1ok) ? b1 : 0.0f;
    acc0 = __builtin_amdgcn_wmma_f32_16x16x4_f32(false, a, false, bf,
                                                 (short)0, acc0, false, false);
    if (doN1){
      float c0 = Bp1[kc0*ldb], c1 = Bp1[kc1*ldb];
      v2f bg; bg.x = (colOK1 && k0ok) ? c0 : 0.0f;
              bg.y = (colOK1 && k1ok) ? c1 : 0.0f;
      acc1 = __builtin_amdgcn_wmma_f32_16x16x4_f32(false, a, false, bg,
                                                   (short)0, acc1, false, false);
    }
  }

  auto epilog = [&](v8f& acc, int colX, bool colOKX){
#pragma unroll
    for (int r = 0; r < 8; ++r){
      int rr = m0 + r + laneHi*8;
      if (rr < M && colOKX){
        float v = acc[r];
        long idx = (long)rr*ldc + colX;
        if (bias)  v += bias[colX];
        if (scale) v = v*scale[colX] + shift[colX];
        if (relu)  v = fmaxf(v, 0.0f);
        if (Cadd)  v += Cab[idx];
        Cb[idx] = v;
      }
    }
  };
  epilog(acc0, col0, colOK0);
  if (doN1) epilog(acc1, col1, colOK1);
}

// ---------------------------------------------------------------- launch
extern "C" void kernel_launch(void* const* d_in, const int* in_sizes, int n_in,
                              void* d_out, int out_size, void* d_ws, size_t ws_size,
                              hipStream_t stream)
{
  const float* x = (const float*)d_in[0];
  const int* ei  = (const int*)d_in[1];
  long E = (long)in_sizes[1] / 2;
  const int* src = ei;
  const int* dst = ei + E;

  int pi = 3;
  const float *cW1[5], *cb1[5], *cg[5], *cbe[5], *crm[5], *crv[5], *cW2[5], *cb2[5];
  for (int i = 0; i < 5; ++i){
    cW1[i]=(const float*)d_in[pi++]; cb1[i]=(const float*)d_in[pi++];
    cg [i]=(const float*)d_in[pi++]; cbe[i]=(const float*)d_in[pi++];
    crm[i]=(const float*)d_in[pi++]; crv[i]=(const float*)d_in[pi++];
    cW2[i]=(const float*)d_in[pi++]; cb2[i]=(const float*)d_in[pi++];
  }
  const float* p1W=(const float*)d_in[pi++]; const float* p1b=(const float*)d_in[pi++];
  const float* p2W=(const float*)d_in[pi++]; const float* p2b=(const float*)d_in[pi++];
  const float* l1W=(const float*)d_in[pi++]; const float* l1b=(const float*)d_in[pi++];
  const float* l1g=(const float*)d_in[pi++]; const float* l1be=(const float*)d_in[pi++];
  const float* l1rm=(const float*)d_in[pi++]; const float* l1rv=(const float*)d_in[pi++];
  const float* l2W=(const float*)d_in[pi++]; const float* l2b=(const float*)d_in[pi++];
  float* outF = (float*)d_out;

  // ---------------- workspace layout
  float* W = (float*)d_ws;
  size_t off = 0;
  auto alloc = [&](size_t n){ float* p = W + off; off += n; return p; };
  const long NODE = (long)B_*N_;
  float* BUF0 = alloc((size_t)NODE*64);
  float* BUF1 = alloc((size_t)NODE*64);
  float* BUF2 = alloc((size_t)NODE*64);
  float* ADJ  = alloc((size_t)B_*N_*N_);
  float* DEG  = alloc((size_t)B_*N_);
  float* S1   = alloc((size_t)B_*N_*K1_);
  float* AS1  = alloc((size_t)B_*N_*K1_);
  float* OADJ1= alloc((size_t)B_*K1_*K1_);
  float* DD1  = alloc((size_t)B_*K1_);
  float* Hm   = alloc(B_*64);
  float* H2   = alloc(B_*64);
  float* BNS  = alloc(6*128);
  // reuse of dead regions
  float* SS1 = AS1;
  float* P0 = BUF0; float* P1 = BUF1; float* P2 = BUF2;
  float* S2    = ADJ;
  float* AS2   = S2 + (size_t)B_*K1_*K2_;
  float* OADJ2 = AS2 + (size_t)B_*K1_*K2_;
  float* DD2   = OADJ2 + (size_t)B_*K2_*K2_;
  float* DEG2  = DD2 + (size_t)B_*K2_;
  float* Q0    = DEG2 + (size_t)B_*K1_;
  float* Q1    = Q0 + (size_t)B_*K2_*64;
  float* Q2    = Q1 + (size_t)B_*K2_*64;
  float* SS2   = AS2;

  auto gs = [](long n){ long b = CDIV(n, 256); if (b > 16384) b = 16384; return dim3((unsigned)b); };
  auto gemm = [&](const float* A, const float* Bm, float* C, const float* Cadd,
                  const float* bias, const float* scale, const float* shift,
                  int relu, int transA, int M, int Nc, int K,
                  int lda, int ldb, int ldc, long sA, long sB, long sC, int batches){
    dim3 grid(CDIV(M,64), CDIV(Nc,32), batches);
    k_gemm<<<grid, 128, 0, stream>>>(A,Bm,C,Cadd,bias,scale,shift,relu,transA,
                                     M,Nc,K,lda,ldb,ldc,sA,sB,sC);
  };

  // loss scalars must start at zero each call
  k_zero<<<1, 64, 0, stream>>>(outF + 896, 2);

  // BN scale/shift precompute: conv1..5 then lin1
  for (int i = 0; i < 5; ++i)
    k_bn_prep<<<1,64,0,stream>>>(cg[i],cbe[i],crm[i],crv[i], BNS+i*128, BNS+i*128+64, 64);
  k_bn_prep<<<1,64,0,stream>>>(l1g,l1be,l1rm,l1rv, BNS+5*128, BNS+5*128+64, 64);

  // ---------------- conv1 (GIN on nodes)
  k_copy<<<gs(NODE*64),256,0,stream>>>(x, BUF0, NODE*64);
  k_scatter<<<8192,256,0,stream>>>(x, BUF0, src, dst, E);
  gemm(BUF0, cW1[0], BUF1, nullptr, cb1[0], BNS+0*128, BNS+0*128+64, 1, 0,
       (int)NODE, 64, 64, 64,64,64, 0,0,0, 1);
  gemm(BUF1, cW2[0], BUF2, nullptr, cb2[0], nullptr, nullptr, 1, 0,
       (int)NODE, 64, 64, 64,64,64, 0,0,0, 1);
  // ---------------- conv2
  k_copy<<<gs(NODE*64),256,0,stream>>>(BUF2, BUF0, NODE*64);
  k_scatter<<<8192,256,0,stream>>>(BUF2, BUF0, src, dst, E);
  gemm(BUF0, cW1[1], BUF1, nullptr, cb1[1], BNS+1*128, BNS+1*128+64, 1, 0,
       (int)NODE, 64, 64, 64,64,64, 0,0,0, 1);
  gemm(BUF1, cW2[1], BUF2, nullptr, cb2[1], nullptr, nullptr, 1, 0,
       (int)NODE, 64, 64, 64,64,64, 0,0,0, 1);      // x2 = BUF2

  // ---------------- dense adjacency + degree
  k_zero<<<gs((long)B_*N_*N_),256,0,stream>>>(ADJ, (long)B_*N_*N_);
  k_adj<<<8192,256,0,stream>>>(ADJ, src, dst, E, N_);
  k_rowsum<<<gs((long)B_*N_),256,0,stream>>>(ADJ, DEG, N_, (long)B_*N_);

  // ---------------- pool1: s = softmax(x2 @ W + b)
  gemm(BUF2, p1W, S1, nullptr, p1b, nullptr, nullptr, 0, 0,
       (int)NODE, K1_, 64, 64, K1_, K1_, 0,0,0, 1);
  k_softmax<<<CDIV(NODE,8),256,0,stream>>>(S1, NODE, K1_);
  // xp1 = s1^T x2
  gemm(S1, BUF2, P0, nullptr, nullptr, nullptr, nullptr, 0, 1,
       K1_, 64, N_, K1_, 64, 64, (long)N_*K1_, (long)N_*64, (long)K1_*64, B_);
  // as1 = adj @ s1
  gemm(ADJ, S1, AS1, nullptr, nullptr, nullptr, nullptr, 0, 0,
       N_, K1_, N_, N_, K1_, K1_, (long)N_*N_, (long)N_*K1_, (long)N_*K1_, B_);
  // out_adj1 = s1^T as1
  gemm(S1, AS1, OADJ1, nullptr, nullptr, nullptr, nullptr, 0, 1,
       K1_, K1_, N_, K1_, K1_, K1_, (long)N_*K1_, (long)N_*K1_, (long)K1_*K1_, B_);
  k_mincut<<<B_,256,0,stream>>>(OADJ1, S1, DEG, N_, K1_, outF+896, 1.0f/B_);
  // ss1 = s1^T s1 (reuses AS1)
  gemm(S1, S1, SS1, nullptr, nullptr, nullptr, nullptr, 0, 1,
       K1_, K1_, N_, K1_, K1_, K1_, (long)N_*K1_, (long)N_*K1_, (long)K1_*K1_, B_);
  k_ortho<<<B_,256,0,stream>>>(SS1, K1_, outF+897, 1.0f/B_);
  k_diag_dd<<<B_,256,0,stream>>>(OADJ1, DD1, K1_);
  k_normalize<<<gs((long)B_*K1_*K1_),256,0,stream>>>(OADJ1, DD1, K1_, (long)B_*K1_*K1_);

  // ---------------- conv3 (dense GIN), rows = B*K1
  gemm(OADJ1, P0, P1, P0, nullptr, nullptr, nullptr, 0, 0,
       K1_, 64, K1_, K1_, 64, 64, (long)K1_*K1_, (long)K1_*64, (long)K1_*64, B_);
  gemm(P1, cW1[2], P2, nullptr, cb1[2], BNS+2*128, BNS+2*128+64, 1, 0,
       B_*K1_, 64, 64, 64,64,64, 0,0,0, 1);
  gemm(P2, cW2[2], P1, nullptr, cb2[2], nullptr, nullptr, 1, 0,
       B_*K1_, 64, 64, 64,64,64, 0,0,0, 1);
  // ---------------- conv4
  gemm(OADJ1, P1, P2, P1, nullptr, nullptr, nullptr, 0, 0,
       K1_, 64, K1_, K1_, 64, 64, (long)K1_*K1_, (long)K1_*64, (long)K1_*64, B_);
  gemm(P2, cW1[3], P0, nullptr, cb1[3], BNS+3*128, BNS+3*128+64, 1, 0,
       B_*K1_, 64, 64, 64,64,64, 0,0,0, 1);
  gemm(P0, cW2[3], P2, nullptr, cb2[3], nullptr, nullptr, 1, 0,
       B_*K1_, 64, 64, 64,64,64, 0,0,0, 1);          // xd(level1) = P2

  // ---------------- pool2
  gemm(P2, p2W, S2, nullptr, p2b, nullptr, nullptr, 0, 0,
       B_*K1_, K2_, 64, 64, K2_, K2_, 0,0,0, 1);
  k_softmax<<<CDIV((long)B_*K1_,8),256,0,stream>>>(S2, (long)B_*K1_, K2_);
  // xp2 = s2^T xd
  gemm(S2, P2, Q0, nullptr, nullptr, nullptr, nullptr, 0, 1,
       K2_, 64, K1_, K2_, 64, 64, (long)K1_*K2_, (long)K1_*64, (long)K2_*64, B_);
  // deg2 from normalized out_adj1
  k_rowsum<<<gs((long)B_*K1_),256,0,stream>>>(OADJ1, DEG2, K1_, (long)B_*K1_);
  // as2 = out_adj1n @ s2
  gemm(OADJ1, S2, AS2, nullptr, nullptr, nullptr, nullptr, 0, 0,
       K1_, K2_, K1_, K1_, K2_, K2_, (long)K1_*K1_, (long)K1_*K2_, (long)K1_*K2_, B_);
  // out_adj2 = s2^T as2
  gemm(S2, AS2, OADJ2, nullptr, nullptr, nullptr, nullptr, 0, 1,
       K2_, K2_, K1_, K2_, K2_, K2_, (long)K1_*K2_, (long)K1_*K2_, (long)K2_*K2_, B_);
  k_mincut<<<B_,256,0,stream>>>(OADJ2, S2, DEG2, K1_, K2_, outF+896, 1.0f/B_);
  // ss2 (reuses AS2)
  gemm(S2, S2, SS2, nullptr, nullptr, nullptr, nullptr, 0, 1,
       K2_, K2_, K1_, K2_, K2_, K2_, (long)K1_*K2_, (long)K1_*K2_, (long)K2_*K2_, B_);
  k_ortho<<<B_,256,0,stream>>>(SS2, K2_, outF+897, 1.0f/B_);
  k_diag_dd<<<B_,256,0,stream>>>(OADJ2, DD2, K2_);
  k_normalize<<<gs((long)B_*K2_*K2_),256,0,stream>>>(OADJ2, DD2, K2_, (long)B_*K2_*K2_);

  // ---------------- conv5
  gemm(OADJ2, Q0, Q1, Q0, nullptr, nullptr, nullptr, 0, 0,
       K2_, 64, K2_, K2_, 64, 64, (long)K2_*K2_, (long)K2_*64, (long)K2_*64, B_);
  gemm(Q1, cW1[4], Q2, nullptr, cb1[4], BNS+4*128, BNS+4*128+64, 1, 0,
       B_*K2_, 64, 64, 64,64,64, 0,0,0, 1);
  gemm(Q2, cW2[4], Q1, nullptr, cb2[4], nullptr, nullptr, 1, 0,
       B_*K2_, 64, 64, 64,64,64, 0,0,0, 1);

  // ---------------- readout head
  k_mean<<<CDIV(B_*64,256),256,0,stream>>>(Q1, Hm, K2_);
  gemm(Hm, l1W, H2, nullptr, l1b, BNS+5*128, BNS+5*128+64, 1, 0,
       B_, 64, 64, 64,64,64, 0,0,0, 1);
  gemm(H2, l2W, outF, nullptr, l2b, nullptr, nullptr, 0, 0,
       B_, 7, 64, 64, 7, 7, 0,0,0, 1);
}